// Conv2d_39273180955611
// MI455X (gfx1250) — compile-verified
//
#include <hip/hip_runtime.h>

// ext-vector types for WMMA operands
typedef __attribute__((ext_vector_type(2))) float v2f;
typedef __attribute__((ext_vector_type(8))) float v8f;

#define CIN   16
#define COUT  16
#define HDIM  64
#define WDIM  64
#define LDIM  512
#define CISTRIDE ((size_t)HDIM * WDIM * LDIM)   // per-input-channel stride in floats

// One block = one (h,w) spatial position. 256 threads = 8 waves (wave32).
// Each wave computes 4 WMMA output tiles of 16(co) x 16(l), covering l0..l0+63.
// K = 144 is walked as 9 taps x 16 ci; each tap = 4 WMMA k-steps of K=4.
__global__ __launch_bounds__(256, 2)
void conv2d_lwe_wmma_f32(const float* __restrict__ x,
                         const float* __restrict__ wgt,
                         const float* __restrict__ bias,
                         float* __restrict__ y)
{
    // Weights permuted to Wlds[(tap*16 + ci)*16 + co]
    __shared__ float Wlds[144 * 16];

    const int tid = threadIdx.x;
    for (int idx = tid; idx < 144 * 16; idx += 256) {
        const int co = idx & 15;
        const int k  = idx >> 4;       // tap*16 + ci
        const int ci = k & 15;
        const int t  = k >> 4;         // tap = kh*3 + kw
        // source: wgt[co][ci][kh][kw] flat = (co*16+ci)*9 + t
        Wlds[idx] = wgt[(co * 16 + ci) * 9 + t];
    }
    __syncthreads();

    const int pos = blockIdx.x;        // 0..4095
    const int h   = pos >> 6;
    const int w   = pos & 63;

    const int lane   = tid & 31;
    const int wave   = tid >> 5;       // 0..7
    const int l15    = lane & 15;      // N index within tile
    const int laneHi = lane >> 4;      // selects K-half (A/B) and M-half (C/D)

    // Bias per accumulator row r: co = r + 8*laneHi
    float brow[8];
    #pragma unroll
    for (int r = 0; r < 8; ++r) brow[r] = bias[r + 8 * laneHi];

    v8f acc[4];
    #pragma unroll
    for (int tile = 0; tile < 4; ++tile) {
        #pragma unroll
        for (int r = 0; r < 8; ++r) acc[tile][r] = 0.0f;
    }

    const int l0 = wave * 64;          // wave covers l0 .. l0+63 (4 tiles of 16)

    #pragma unroll
    for (int t = 0; t < 9; ++t) {
        const int kh = t / 3, kw = t % 3;
        const int hh = h + kh - 1;
        const int ww = w + kw - 1;
        // Uniform (scalar) predicate: zero-padding taps contribute nothing.
        if ((unsigned)hh < (unsigned)HDIM && (unsigned)ww < (unsigned)WDIM) {
            const float* tap_base = x + ((size_t)hh * WDIM + ww) * LDIM;

            // A fragments for this tap: a[u][j] holds W[m = l15, k = t*16 + 4u + 2*laneHi + j]
            v2f a[4];
            #pragma unroll
            for (int u = 0; u < 4; ++u) {
                #pragma unroll
                for (int j = 0; j < 2; ++j) {
                    a[u][j] = Wlds[(t * 16 + 4 * u + 2 * laneHi + j) * 16 + l15];
                }
            }

            #pragma unroll
            for (int u = 0; u < 4; ++u) {
                const int ciBase = 4 * u + 2 * laneHi;     // per-lane ci for j=0
                const float* p0 = tap_base + (size_t)ciBase * CISTRIDE + l0 + l15;
                const float* p1 = p0 + CISTRIDE;           // j=1 -> ci+1
                #pragma unroll
                for (int tile = 0; tile < 4; ++tile) {
                    v2f b;
                    b[0] = p0[tile * 16];
                    b[1] = p1[tile * 16];
                    acc[tile] = __builtin_amdgcn_wmma_f32_16x16x4_f32(
                        /*neg_a=*/false, a[u],
                        /*neg_b=*/false, b,
                        /*c_mod=*/(short)0, acc[tile],
                        /*reuse_a=*/false, /*reuse_b=*/false);
                }
            }
        }
    }

    // Store: y[((co*64 + h)*64 + w)*512 + l], co = r + 8*laneHi, l = l0 + tile*16 + l15
    #pragma unroll
    for (int tile = 0; tile < 4; ++tile) {
        #pragma unroll
        for (int r = 0; r < 8; ++r) {
            const int co = r + 8 * laneHi;
            const size_t oi = (((size_t)co * HDIM + h) * WDIM + w) * LDIM
                            + (size_t)(l0 + tile * 16 + l15);
            y[oi] = acc[tile][r] + brow[r];
        }
    }
}

extern "C" void kernel_launch(void* const* d_in, const int* in_sizes, int n_in,
                              void* d_out, int out_size, void* d_ws, size_t ws_size,
                              hipStream_t stream) {
    (void)in_sizes; (void)n_in; (void)out_size; (void)d_ws; (void)ws_size;
    const float* x    = (const float*)d_in[0];   // (1,16,64,64,512)
    const float* wgt  = (const float*)d_in[1];   // (16,16,3,3)
    const float* bias = (const float*)d_in[2];   // (16,)
    float* y = (float*)d_out;                    // (1,16,64,64,512)

    dim3 grid(HDIM * WDIM);   // 4096 blocks, one per (h,w)
    dim3 block(256);          // 8 waves
    conv2d_lwe_wmma_f32<<<grid, block, 0, stream>>>(x, wgt, bias, y);
}